// gPool_8632884265215
// MI455X (gfx1250) — compile-verified
//
#include <hip/hip_runtime.h>

typedef __attribute__((ext_vector_type(2))) float v2f;
typedef __attribute__((ext_vector_type(8))) float v8f;

#define N_NODES  100000
#define D_FEAT   256
#define N_EDGES  3200000
#define TOP_K    50000

// ---- radix sort config: 49-bit keys, 10-bit digits, 5 passes ----
#define RS_BITS    10
#define RS_RADIX   1024
#define RS_PASSES  5
#define RS_EPB     256                                   // elements per block (1 wave x 8 rounds)
#define RS_ROUNDS  (RS_EPB / 32)
#define RS_BLOCKS  ((N_NODES + RS_EPB - 1) / RS_EPB)     // 391
#define RS_ENTRIES (RS_RADIX * RS_BLOCKS)

// ------------------------------------------------------------------
// scores = sigmoid(h @ W + b) via V_WMMA_F32_16X16X4_F32 (fp32-exact).
// One wave per 16-row tile. A: 16x4 f32 slab of h (lanes 0-15 -> K=0,1;
// lanes 16-31 -> K=2,3). B: 4x16 with W[k..k+3] replicated to all 16
// columns -> every column of D equals the 16 scores.
// Emits sortable key: (~bits(score) << 17) | row  (49 bits, unique).
// ------------------------------------------------------------------
__global__ void score_wmma(const float* __restrict__ h,
                           const float* __restrict__ W,
                           const float* __restrict__ bias,
                           unsigned long long* __restrict__ keys) {
  const int lane    = threadIdx.x;            // 0..31
  const int rowBase = blockIdx.x * 16;
  const int m       = lane & 15;
  const int koff    = (lane >> 4) * 2;        // upper half-wave holds K+2,K+3
  const float* hrow = h + (size_t)(rowBase + m) * D_FEAT;

  v8f acc = {0.f, 0.f, 0.f, 0.f, 0.f, 0.f, 0.f, 0.f};
  #pragma unroll 4
  for (int k = 0; k < D_FEAT; k += 4) {
    v2f a  = *(const v2f*)(hrow + k + koff);
    v2f bv = *(const v2f*)(W + k + koff);
    acc = __builtin_amdgcn_wmma_f32_16x16x4_f32(false, a, false, bv,
                                                (short)0, acc, false, false);
  }

  if ((lane & 15) == 0) {                     // lanes 0 and 16: columns N=0
    const float bz  = bias[0];
    const int  half = (lane >> 4) * 8;        // D layout: VGPR j -> M=j / M=8+j
    #pragma unroll
    for (int j = 0; j < 8; ++j) {
      int   row = rowBase + half + j;
      float z   = acc[j] + bz;
      float s   = __builtin_amdgcn_rcpf(1.0f + __expf(-z));  // fast sigmoid (v_rcp_f32)
      unsigned bits = __float_as_uint(s);     // sigmoid>0 -> bits monotone in s
      keys[row] = ((unsigned long long)(~bits) << 17) | (unsigned)row;
    }
  }
}

// ---------------- radix sort: per-block digit histogram ----------------
__global__ void radix_hist(const unsigned long long* __restrict__ keys,
                           int shift, unsigned* __restrict__ hist) {
  __shared__ unsigned lh[RS_RADIX];
  const int lane = threadIdx.x;
  for (int i = lane; i < RS_RADIX; i += 32) lh[i] = 0;
  __syncthreads();
  const int base = blockIdx.x * RS_EPB;
  for (int r = 0; r < RS_ROUNDS; ++r) {
    int i = base + r * 32 + lane;
    if (i < N_NODES) {
      unsigned d = (unsigned)(keys[i] >> shift) & (RS_RADIX - 1);
      atomicAdd(&lh[d], 1u);
    }
  }
  __syncthreads();
  for (int d = lane; d < RS_RADIX; d += 32) {
    unsigned c = lh[d];
    if (c) hist[(size_t)d * RS_BLOCKS + blockIdx.x] = c;   // hist pre-zeroed
  }
}

// ------------- exclusive scan over digit-major histogram ---------------
__global__ void radix_scan(unsigned* __restrict__ hist) {
  __shared__ unsigned tmp[1024];
  __shared__ unsigned carry;
  const int t = threadIdx.x;
  if (t == 0) carry = 0;
  __syncthreads();
  for (int base = 0; base < RS_ENTRIES; base += 1024) {
    unsigned v = (base + t < RS_ENTRIES) ? hist[base + t] : 0u;
    tmp[t] = v;
    __syncthreads();
    for (int off = 1; off < 1024; off <<= 1) {
      unsigned add = (t >= off) ? tmp[t - off] : 0u;
      __syncthreads();
      tmp[t] += add;
      __syncthreads();
    }
    unsigned excl = tmp[t] - v + carry;
    if (base + t < RS_ENTRIES) hist[base + t] = excl;
    __syncthreads();
    if (t == 0) carry += tmp[1023];
    __syncthreads();
  }
}

// ------- stable scatter: wave32 ballot multisplit + LDS counters -------
__global__ void radix_scatter(const unsigned long long* __restrict__ src,
                              unsigned long long* __restrict__ dst,
                              int shift, const unsigned* __restrict__ hist) {
  __shared__ unsigned cnt[RS_RADIX];
  const int lane = threadIdx.x;
  for (int i = lane; i < RS_RADIX; i += 32) cnt[i] = 0;
  __syncthreads();
  const int base = blockIdx.x * RS_EPB;
  const unsigned lt = (1u << lane) - 1u;
  for (int r = 0; r < RS_ROUNDS; ++r) {
    int  i     = base + r * 32 + lane;
    bool valid = (i < N_NODES);
    unsigned long long key = valid ? src[i] : 0ull;
    unsigned d = (unsigned)(key >> shift) & (RS_RADIX - 1);
    unsigned peers = (unsigned)__ballot(valid);
    #pragma unroll
    for (int bit = 0; bit < RS_BITS; ++bit) {
      unsigned vote = (unsigned)__ballot(valid && ((d >> bit) & 1u));
      peers &= ((d >> bit) & 1u) ? vote : ~vote;
    }
    if (valid) {
      unsigned rank   = __popc(peers & lt);           // stable in-round rank
      unsigned before = cnt[d];                       // earlier rounds, this block
      unsigned gbase  = hist[(size_t)d * RS_BLOCKS + blockIdx.x];
      dst[gbase + before + rank] = key;
      if (rank == 0) cnt[d] = before + __popc(peers); // one leader per digit
    }
    __syncthreads();
  }
}

// ------------------------- epilogue kernels ----------------------------
__global__ void build_member(const unsigned long long* __restrict__ sorted,
                             unsigned* __restrict__ member) {
  int r = blockIdx.x * blockDim.x + threadIdx.x;
  if (r < TOP_K) {
    unsigned idx = (unsigned)(sorted[r] & 0x1FFFFu);
    atomicOr(&member[idx >> 5], 1u << (idx & 31u));
  }
}

// new_h[r] = h[idx] * score, vectorized float4 (global_load/store_b128).
__global__ void gather_scale(const unsigned long long* __restrict__ sorted,
                             const float* __restrict__ h,
                             float* __restrict__ out) {
  int r = blockIdx.x;
  unsigned long long key = sorted[r];
  unsigned idx = (unsigned)(key & 0x1FFFFu);
  float    s   = __uint_as_float(~(unsigned)(key >> 17));  // exact score bits
  int c4 = threadIdx.x;                                    // 0..63, 4 floats each
  const float4* src = (const float4*)(h + (size_t)idx * D_FEAT);
  float4*       dst = (float4*)(out + (size_t)r * D_FEAT);
  float4 v = src[c4];
  v.x *= s; v.y *= s; v.z *= s; v.w *= s;
  dst[c4] = v;
  if (c4 == 0) out[(size_t)TOP_K * D_FEAT + r] = (float)idx; // node_ids section
}

__global__ void edge_mask_kernel(const int* __restrict__ ei,
                                 const unsigned* __restrict__ member,
                                 float* __restrict__ out) {
  int e = blockIdx.x * blockDim.x + threadIdx.x;
  if (e < N_EDGES) {
    unsigned s = (unsigned)ei[e];
    unsigned d = (unsigned)ei[N_EDGES + e];
    bool ms = (member[s >> 5] >> (s & 31u)) & 1u;
    bool md = (member[d >> 5] >> (d & 31u)) & 1u;
    out[e] = (ms && md) ? 1.0f : 0.0f;
  }
}

// -----------------------------------------------------------------------
extern "C" void kernel_launch(void* const* d_in, const int* in_sizes, int n_in,
                              void* d_out, int out_size, void* d_ws, size_t ws_size,
                              hipStream_t stream) {
  (void)in_sizes; (void)n_in; (void)out_size; (void)ws_size;
  const float* h  = (const float*)d_in[0];
  const float* W  = (const float*)d_in[1];
  const float* b  = (const float*)d_in[2];
  const int*   ei = (const int*)d_in[3];
  float* out = (float*)d_out;

  // workspace layout
  char* ws = (char*)d_ws;
  unsigned long long* keysA  = (unsigned long long*)ws;          // 800 KB
  unsigned long long* keysB  = keysA + N_NODES;                  // 800 KB
  unsigned*           hist   = (unsigned*)(keysB + N_NODES);     // ~1.6 MB
  unsigned*           member = hist + RS_ENTRIES;                // 12.5 KB

  hipMemsetAsync(member, 0, ((N_NODES + 31) / 32) * sizeof(unsigned), stream);

  // 1) fp32 WMMA GEMV -> sigmoid -> sortable 49-bit keys
  score_wmma<<<N_NODES / 16, 32, 0, stream>>>(h, W, b, keysA);

  // 2) exact descending top-k order: 5-pass LSD radix sort (stable)
  unsigned long long* src = keysA;
  unsigned long long* dst = keysB;
  for (int p = 0; p < RS_PASSES; ++p) {
    int shift = p * RS_BITS;
    hipMemsetAsync(hist, 0, (size_t)RS_ENTRIES * sizeof(unsigned), stream);
    radix_hist   <<<RS_BLOCKS, 32, 0, stream>>>(src, shift, hist);
    radix_scan   <<<1, 1024,   0, stream>>>(hist);
    radix_scatter<<<RS_BLOCKS, 32, 0, stream>>>(src, dst, shift, hist);
    unsigned long long* t = src; src = dst; dst = t;
  }
  const unsigned long long* sorted = src;   // ends in keysB after 5 passes

  // 3) membership bitset, gathered+scaled features, node ids, edge mask
  build_member<<<(TOP_K + 255) / 256, 256, 0, stream>>>(sorted, member);
  gather_scale<<<TOP_K, D_FEAT / 4, 0, stream>>>(sorted, h, out);
  edge_mask_kernel<<<(N_EDGES + 255) / 256, 256, 0, stream>>>(
      ei, member, out + (size_t)TOP_K * D_FEAT + TOP_K);
}